// eulercell_91156385890726
// MI455X (gfx1250) — compile-verified
//
#include <hip/hip_runtime.h>

typedef __attribute__((ext_vector_type(2))) float v2f;
typedef __attribute__((ext_vector_type(8))) float v8f;

#define NSTEPS 10
#define TSTEP  100.0f
#define F0C    10.0f
#define OMEGAC 30.0f

__device__ __forceinline__ float bcast0(float v) {
    // lanes 0-15 all hold the same value (B was column-replicated); broadcast to whole wave
    return __int_as_float(__builtin_amdgcn_readfirstlane(__float_as_int(v)));
}

__global__ __launch_bounds__(32)
void eulercell_91156385890726_kernel(const float* __restrict__ Yin,
                                     const float* __restrict__ act_acc,
                                     const float* __restrict__ W,
                                     const float* __restrict__ bvec,
                                     const float* __restrict__ pk1,
                                     const float* __restrict__ pk2,
                                     const float* __restrict__ pk3,
                                     float* __restrict__ out) {
    const int lane = threadIdx.x;
    const int m    = lane & 15;
    const int hi   = lane >> 4;

    const float k1 = pk1[0], k2 = pk2[0], k3 = pk3[0];
    // K = [[k1+k2, -k2, 0], [-k2, k2+k3, -k3], [0, -k3, k3]]
    const float K00 = k1 + k2, K01 = -k2;
    const float K10 = -k2,     K11 = k2 + k3, K12 = -k3;
    const float K21 = -k3,     K22 = k3;
    const float Mi0 = 0.1f, Mi1 = 0.02f, Mi2 = 0.05f;   // 1/diag(M)

    // --- A tiles for Yn = W @ p  (K split 4 + 2), ISA 16x4 f32 A layout:
    // lane<16: v0=K0 v1=K1 ; lane>=16: v0=K2 v1=K3 (row M = lane&15)
    const bool rowv = (m < 6);
    const int  c0   = hi ? 2 : 0;
    v2f a1, a2;
    a1.x = rowv ? W[m * 6 + c0]     : 0.0f;
    a1.y = rowv ? W[m * 6 + c0 + 1] : 0.0f;
    a2.x = (rowv && !hi) ? W[m * 6 + 4] : 0.0f;   // K-slots 2,3 of 2nd tile are zero
    a2.y = (rowv && !hi) ? W[m * 6 + 5] : 0.0f;

    const float b0 = bvec[0], b1 = bvec[1], b2 = bvec[2];
    const float b3 = bvec[3], b4 = bvec[4], b5 = bvec[5];

    float y0 = Yin[0], y1 = Yin[1], y2 = Yin[2];
    float y3 = Yin[3], y4 = Yin[4], y5 = Yin[5];

    float flast = 0.0f;
#pragma unroll
    for (int t = 0; t < NSTEPS; ++t) {
        const float f = F0C * sinf(OMEGAC * (float)t);
        flast = f;

        // p = Y + dt * A_inv @ (F1 - B @ Y^T)
        const float Kx0 = K00 * y3 + K01 * y4;
        const float Kx1 = K10 * y3 + K11 * y4 + K12 * y5;
        const float Kx2 = K21 * y4 + K22 * y5;
        const float p0 = y0 + TSTEP * ((0.0f - Kx0) * Mi0);
        const float p1 = y1 + TSTEP * ((f    - Kx1) * Mi1);
        const float p2 = y2 + TSTEP * ((0.0f - Kx2) * Mi2);
        const float p3 = y3 + TSTEP * y0;
        const float p4 = y4 + TSTEP * y1;
        const float p5 = y5 + TSTEP * y2;

        // B tiles (4x16 f32), every column = p  -> every D column = W@p.
        // lane<16: v0=K0 v1=K1 ; lane>=16: v0=K2 v1=K3
        v2f bm1, bm2;
        bm1.x = hi ? p2 : p0;
        bm1.y = hi ? p3 : p1;
        bm2.x = hi ? 0.0f : p4;
        bm2.y = hi ? 0.0f : p5;

        v8f c = {};
        c = __builtin_amdgcn_wmma_f32_16x16x4_f32(false, a1, false, bm1,
                                                  (short)0, c, false, false);
        c = __builtin_amdgcn_wmma_f32_16x16x4_f32(false, a2, false, bm2,
                                                  (short)0, c, false, false);

        // D row i (=Yn_matmul[i]) lives in VGPR i of lanes 0-15; broadcast + bias.
        y0 = bcast0(c[0]) + b0;
        y1 = bcast0(c[1]) + b1;
        y2 = bcast0(c[2]) + b2;
        y3 = bcast0(c[3]) + b3;
        y4 = bcast0(c[4]) + b4;
        y5 = bcast0(c[5]) + b5;

        // acc = Minv @ (F1[:3] - K @ X), X = Yn[3:6]
        const float Ka0 = K00 * y3 + K01 * y4;
        const float Ka1 = K10 * y3 + K11 * y4 + K12 * y5;
        const float Ka2 = K21 * y4 + K22 * y5;
        const float acc0 = (0.0f - Ka0) * Mi0;
        const float acc1 = (f    - Ka1) * Mi1;
        const float acc2 = (0.0f - Ka2) * Mi2;

        if (lane == 0) {
            out[15 + 3 * t + 0] = y3;   // DISP
            out[15 + 3 * t + 1] = y4;
            out[15 + 3 * t + 2] = y5;
            out[45 + 3 * t + 0] = acc0; // ACC
            out[45 + 3 * t + 1] = acc1;
            out[45 + 3 * t + 2] = acc2;
        }
    }

    if (lane == 0) {
        // Yf
        out[0] = y0; out[1] = y1; out[2] = y2;
        out[3] = y3; out[4] = y4; out[5] = y5;
        // pred_k = K (row-major)
        out[6]  = K00; out[7]  = K01; out[8]  = 0.0f;
        out[9]  = K10; out[10] = K11; out[11] = K12;
        out[12] = 0.0f; out[13] = K21; out[14] = K22;
        // l1 = M @ act_acc + K @ Yf[3:6] - F1_last[:3]
        out[75] = 10.0f * act_acc[0] + (K00 * y3 + K01 * y4);
        out[76] = 50.0f * act_acc[1] + (K10 * y3 + K11 * y4 + K12 * y5) - flast;
        out[77] = 20.0f * act_acc[2] + (K21 * y4 + K22 * y5);
    }
}

extern "C" void kernel_launch(void* const* d_in, const int* in_sizes, int n_in,
                              void* d_out, int out_size, void* d_ws, size_t ws_size,
                              hipStream_t stream) {
    (void)in_sizes; (void)n_in; (void)out_size; (void)d_ws; (void)ws_size;
    eulercell_91156385890726_kernel<<<1, 32, 0, stream>>>(
        (const float*)d_in[0],   // Y (1,6)
        (const float*)d_in[1],   // act_acc (3,1)
        (const float*)d_in[2],   // W (6,6)
        (const float*)d_in[3],   // b (6,)
        (const float*)d_in[4],   // k1
        (const float*)d_in[5],   // k2
        (const float*)d_in[6],   // k3
        (float*)d_out);          // 78 f32
}